// SparseMixerMoeRoutingMethod_66340064854665
// MI455X (gfx1250) — compile-verified
//
#include <hip/hip_runtime.h>
#include <stdint.h>

// SparseMixer MoE routing for MI455X (gfx1250).
// HBM roofline: 160 MiB @ 23.3 TB/s ~= 7 us; working set fits the 192 MB L2,
// so replays become L2/VALU bound -> keep RT caching, minimize VALU.
// Data path:
//   global --(global_load_async_to_lds_b128 saddr-form, ASYNCcnt)--> padded LDS
//   LDS --(ds_load_b128, 80 B/token stride => optimal 2-pass banking)--> VGPRs
//   wave32 VALU + v_exp_f32 (TRANS co-exec) routing math, b64 stores.

#define NEXP 16
#define TPB 256
#define PADF 20                      // floats per token slot in LDS (80 B)
#define LOG2E 1.4426950408889634f

__device__ __forceinline__ void sparsemixer_pass(const float f[NEXP], int exclude,
                                                 int& out_i, float& out_v) {
  // Knock out `exclude` (pass 1 uses -1: compares fold away at compile time).
  float g[NEXP];
#pragma unroll
  for (int i = 0; i < NEXP; ++i)
    g[i] = (i == exclude) ? -__builtin_inff() : f[i];

  // Max via fmax chain (backend fuses exact v_max3_num_f32 triples).
  float m = g[0];
#pragma unroll
  for (int i = 1; i < NEXP; ++i) m = fmaxf(m, g[i]);

  // First index attaining the max (fmax returns an input bit-exactly, so
  // equality is exact; descending selects give jnp.argmax's first-match).
  int idx = NEXP - 1;
#pragma unroll
  for (int i = NEXP - 2; i >= 0; --i) idx = (g[i] == m) ? i : idx;

  // Division-free mask:  rel > 2*eps  <=>  (m - g) > 0.02*max(|g|, m).
  // Edge cases match JAX: g=-inf -> inf>inf false -> kept, exp(-inf)=0;
  // g=m -> kept, contributes 1; g=0,m=0 -> 0>0 false (ref: NaN>eps false).
  float sum = 0.0f;
#pragma unroll
  for (int i = 0; i < NEXP; ++i) {
    float d   = m - g[i];                          // v_sub
    float thr = 0.02f * fmaxf(fabsf(g[i]), m);     // abs is a src modifier
    float e   = __builtin_amdgcn_exp2f(-d * LOG2E);// TRANS, neg src modifier
    sum += (d > thr) ? 0.0f : e;                   // cmp + cndmask + add
  }
  out_i = idx;
  out_v = 1.0f / sum;                              // softmax value at argmax
}

__device__ __forceinline__ void sparsemixer_compute_store(
    const float* tile, int tid, int64_t tok,
    int* __restrict__ out_idx, float* __restrict__ out_val) {
  const float4* row = (const float4*)&tile[tid * PADF];
  float4 r0 = row[0], r1 = row[1], r2 = row[2], r3 = row[3];
  float f[NEXP] = {r0.x, r0.y, r0.z, r0.w, r1.x, r1.y, r1.z, r1.w,
                   r2.x, r2.y, r2.z, r2.w, r3.x, r3.y, r3.z, r3.w};
  int i0, i1;
  float v0, v1;
  sparsemixer_pass(f, -1, i0, v0);   // top-1
  sparsemixer_pass(f, i0, i1, v1);   // top-2 with top-1 knocked out

  int2 oi;   oi.x = i0; oi.y = i1;
  float2 ov; ov.x = v0; ov.y = v1;
  *(int2*)(out_idx + tok * 2) = oi;
  *(float2*)(out_val + tok * 2) = ov;
}

// Guard-free main kernel: one full 256-token tile per block.
extern "C" __global__ void __launch_bounds__(TPB)
sparsemixer_route_full(const float* __restrict__ logits, int* __restrict__ out_idx,
                       float* __restrict__ out_val) {
  __shared__ __align__(16) float tile[TPB * PADF];   // 20 KB
  const int tid = threadIdx.x;
  const uint32_t tok0 = (uint32_t)blockIdx.x * TPB;

  // SADDR-form async DMA: uniform 64-bit base in SGPRs, per-lane 32-bit byte
  // offset. Chunk c = tid + 256k -> contiguous 512 B per wave-instruction on
  // the global side; per-lane LDS dest lands it in the padded 80 B layout.
  const unsigned long long gbase =
      (unsigned long long)(uintptr_t)(logits + (size_t)tok0 * NEXP);
  const uint32_t goff0 = (uint32_t)tid * 16u;                       // chunk tid
  const uint32_t lds0 =
      (uint32_t)(uintptr_t)&tile[(tid >> 2) * PADF + (tid & 3) * 4];
#pragma unroll
  for (int k = 0; k < 4; ++k) {
    // token advances by 64 per k: LDS += 64*80, global += 256*16
    asm volatile("global_load_async_to_lds_b128 %0, %1, %2"
                 :: "v"(lds0 + (uint32_t)(k * 64 * PADF * 4)),
                    "v"(goff0 + (uint32_t)(k * TPB * 16)),
                    "s"(gbase)
                 : "memory");
  }
  asm volatile("s_wait_asynccnt 0" ::: "memory");  // own-wave DMA complete
  __syncthreads();                                 // cross-wave visibility

  sparsemixer_compute_store(tile, tid, (int64_t)tok0 + tid, out_idx, out_val);
}

// Guarded tail kernel (only launched when n_tok % 256 != 0).
extern "C" __global__ void __launch_bounds__(TPB)
sparsemixer_route_tail(const float* __restrict__ logits, int* __restrict__ out_idx,
                       float* __restrict__ out_val, int n_tok, int tok_base) {
  __shared__ __align__(16) float tile[TPB * PADF];
  const int tid = threadIdx.x;
  const int64_t tok0 = tok_base;

#pragma unroll
  for (int k = 0; k < 4; ++k) {
    const int c = tid + TPB * k;
    const int64_t tok = tok0 + (c >> 2);
    if (tok < n_tok) {
      const float* g = logits + tok * NEXP + (c & 3) * 4;
      const uint32_t lds =
          (uint32_t)(uintptr_t)&tile[(c >> 2) * PADF + (c & 3) * 4];
      asm volatile("global_load_async_to_lds_b128 %0, %1, off"
                   :: "v"(lds), "v"(g)
                   : "memory");
    }
  }
  asm volatile("s_wait_asynccnt 0" ::: "memory");
  __syncthreads();

  const int64_t tok = tok0 + tid;
  if (tok >= n_tok) return;
  sparsemixer_compute_store(tile, tid, tok, out_idx, out_val);
}

extern "C" void kernel_launch(void* const* d_in, const int* in_sizes, int n_in,
                              void* d_out, int out_size, void* d_ws, size_t ws_size,
                              hipStream_t stream) {
  const float* logits = (const float*)d_in[0];
  const int n_tok = in_sizes[0] / NEXP;              // 2^21
  // d_out layout: [N*2] int32 indices, then [N*2] f32 values (flat, in order)
  int* out_idx = (int*)d_out;
  float* out_val = (float*)d_out + (size_t)n_tok * 2;

  const int full_blocks = n_tok / TPB;               // 8192 for 2^21 tokens
  const int rem = n_tok - full_blocks * TPB;
  if (full_blocks)
    sparsemixer_route_full<<<full_blocks, TPB, 0, stream>>>(logits, out_idx, out_val);
  if (rem)
    sparsemixer_route_tail<<<1, TPB, 0, stream>>>(logits, out_idx, out_val,
                                                  n_tok, full_blocks * TPB);
}